// YoloLayer_12644383719720
// MI455X (gfx1250) — compile-verified
//
#include <hip/hip_runtime.h>
#include <stdint.h>

#define GHD 64
#define GWD 64
#define NA 3
#define NT 100
#define CELLS_PER_BATCH (GHD * GWD * NA)      // 12288
#define WAVES_PER_BLOCK 8
#define BLOCKS_PER_BATCH 48
#define WAVES_PER_BATCH (WAVES_PER_BLOCK * BLOCKS_PER_BATCH)  // 384 (divisible by 3)

__device__ __forceinline__ float wave_max32(float v) {
#pragma unroll
  for (int s = 16; s > 0; s >>= 1) v = fmaxf(v, __shfl_xor(v, s, 32));
  return v;
}
__device__ __forceinline__ float wave_sum32(float v) {
#pragma unroll
  for (int s = 16; s > 0; s >>= 1) v += __shfl_xor(v, s, 32);
  return v;
}

__global__ void __launch_bounds__(256) yolo_zero_kernel(float* out, int n) {
  int i = threadIdx.x;
  if (i < n) out[i] = 0.0f;
}

__global__ void __launch_bounds__(256) yolo_loss_kernel(
    const float* __restrict__ y_pred, const float* __restrict__ y_true,
    const float* __restrict__ true_boxes, float* __restrict__ out) {
  __shared__ float s_box[NT * 4];  // true_boxes[b], pre-scaled (xy/64, wh/1024)

  const int b = blockIdx.y;
  const int tid = threadIdx.x;
  const int lane = tid & 31;
  const int wave = tid >> 5;

  // ---- Stage true_boxes[b] (100 boxes * 16B) into LDS via CDNA5 async DMA ----
  if (tid < NT) {
    const float* g = true_boxes + (size_t)b * NT * 4 + (size_t)tid * 4;
    // Flat address of a __shared__ object: low 32 bits are the LDS byte offset
    // (shared aperture base lives entirely in addr[63:32] on gfx1250).
    unsigned lds_off = (unsigned)(uintptr_t)(&s_box[tid * 4]);
    asm volatile("global_load_async_to_lds_b128 %0, %1, off"
                 :: "v"(lds_off), "v"(g)
                 : "memory");
  }
  asm volatile("s_wait_asynccnt 0" ::: "memory");
  __syncthreads();
  // Pre-scale once per block: xy -> /GW, wh -> /NET (reused by 256 cells).
  {
    int i = tid;
    if (i < NT * 4) s_box[i] *= ((i & 3) < 2) ? (1.0f / GWD) : (1.0f / 1024.0f);
    i = tid + 256;
    if (i < NT * 4) s_box[i] *= ((i & 3) < 2) ? (1.0f / GWD) : (1.0f / 1024.0f);
  }
  __syncthreads();

  const float inv_gw = 1.0f / GWD;
  const float inv_net = 1.0f / 1024.0f;

  const int wgid = blockIdx.x * WAVES_PER_BLOCK + wave;  // wave id within batch

  // Stride 384 is divisible by 3 -> anchor index is loop-invariant per wave.
  const int a = wgid % 3;
  const float aw_tab[NA] = {116.0f, 156.0f, 373.0f};
  const float ah_tab[NA] = {90.0f, 198.0f, 326.0f};
  const float aw = aw_tab[a] * inv_net;   // pre-fold /NET into anchor
  const float ah = ah_tab[a] * inv_net;

  int r = wgid / 3;                        // (gy*64+gx); advances by 128/iter
  const float* pp = y_pred + ((size_t)b * CELLS_PER_BATCH + wgid) * 85;
  const float* tp = y_true + ((size_t)b * CELLS_PER_BATCH + wgid) * 85;
  const size_t cell_stride = (size_t)WAVES_PER_BATCH * 85;

  float acc = 0.0f;

  for (int cell = wgid; cell < CELLS_PER_BATCH;
       cell += WAVES_PER_BATCH, r += CELLS_PER_BATCH / (NA * 32),  // +128
       pp += cell_stride, tp += cell_stride) {
    const int gx = r & (GWD - 1);
    const int gy = r >> 6;

    // gfx1250 global_prefetch_b8 for the next streamed cell of this wave
    if (cell + WAVES_PER_BATCH < CELLS_PER_BATCH) {
      __builtin_prefetch(pp + cell_stride, 0, 1);
      __builtin_prefetch(tp + cell_stride, 0, 1);
    }

    // 85 elements striped e = lane + 32k, k = 0..2 (fully coalesced b32 loads,
    // 128B per instruction per wave)
    float pr0 = pp[lane], tr0 = tp[lane];
    float pr1 = pp[lane + 32], tr1 = tp[lane + 32];
    float pr2 = (lane < 21) ? pp[lane + 64] : 0.0f;
    float tr2 = (lane < 21) ? tp[lane + 64] : 0.0f;

    // Broadcast the 5 scalar slots (elements 0..4 live in pr0/tr0 of lanes 0..4)
    float p0 = __shfl(pr0, 0, 32), p1 = __shfl(pr0, 1, 32);
    float p2 = __shfl(pr0, 2, 32), p3 = __shfl(pr0, 3, 32);
    float p4 = __shfl(pr0, 4, 32);
    float t0 = __shfl(tr0, 0, 32), t1 = __shfl(tr0, 1, 32);
    float t2 = __shfl(tr0, 2, 32), t3 = __shfl(tr0, 3, 32);
    float t4 = __shfl(tr0, 4, 32);

    // ---- log-softmax over pred class logits (elements 5..84) ----
    float m = -__builtin_inff();
    if (lane >= 5) m = pr0;            // classes 0..26
    m = fmaxf(m, pr1);                 // classes 27..58
    if (lane < 21) m = fmaxf(m, pr2);  // classes 59..79
    m = wave_max32(m);
    float se = 0.0f;
    if (lane >= 5) se += __expf(pr0 - m);
    se += __expf(pr1 - m);
    if (lane < 21) se += __expf(pr2 - m);
    se = wave_sum32(se);
    float lse = m + __logf(se);

    // ---- argmax over true class probs (first-occurrence tie rule) ----
    float bv = -__builtin_inff();
    int bi = 0x7fffffff;
    if (lane >= 5) { bv = tr0; bi = lane - 5; }
    if (tr1 > bv) { bv = tr1; bi = lane + 27; }
    if (lane < 21 && tr2 > bv) { bv = tr2; bi = lane + 59; }
#pragma unroll
    for (int s = 16; s > 0; s >>= 1) {
      float ov = __shfl_xor(bv, s, 32);
      int oi = __shfl_xor(bi, s, 32);
      if (ov > bv || (ov == bv && oi < bi)) { bv = ov; bi = oi; }
    }
    int e = bi + 5;  // wave-uniform element index of the true class logit
    float l0 = __shfl(pr0, e & 31, 32);
    float l1 = __shfl(pr1, e & 31, 32);
    float l2 = __shfl(pr2, e & 31, 32);
    int kreg = e >> 5;
    float logit_t = (kreg == 0) ? l0 : (kreg == 1) ? l1 : l2;
    float ce = lse - logit_t;  // -log_softmax[true_class]

    // ---- predicted box (cell coords + normalized coords for IoU) ----
    float sx = 1.0f / (1.0f + __expf(-p0));
    float sy = 1.0f / (1.0f + __expf(-p1));
    float pbx = (float)gx + sx, pby = (float)gy + sy;
    float pxc = pbx * inv_gw, pyc = pby * inv_gw;
    float pw = __expf(p2) * aw;
    float ph = __expf(p3) * ah;
    float parea = pw * ph;
    float pminx = pxc - 0.5f * pw, pmaxx = pxc + 0.5f * pw;
    float pminy = pyc - 0.5f * ph, pmaxy = pyc + 0.5f * ph;

    // ---- best IoU against 100 staged boxes (lanes own tb = lane + 32k) ----
    float biou = -__builtin_inff();
#pragma unroll
    for (int kk = 0; kk < 4; ++kk) {
      int tb = lane + 32 * kk;
      if (tb < NT) {
        float bx = s_box[4 * tb + 0], by = s_box[4 * tb + 1];
        float bw = s_box[4 * tb + 2], bh = s_box[4 * tb + 3];
        float iw = fmaxf(fminf(pmaxx, bx + 0.5f * bw) - fmaxf(pminx, bx - 0.5f * bw), 0.0f);
        float ih = fmaxf(fminf(pmaxy, by + 0.5f * bh) - fmaxf(pminy, by - 0.5f * bh), 0.0f);
        float inter = iw * ih;
        float uni = parea + bw * bh - inter;
        biou = fmaxf(biou, inter / uni);
      }
    }
    biou = wave_max32(biou);
    float ign = (biou > 0.5f) ? 1.0f : 0.0f;

    // ---- deltas ----
    float obj = t4;
    float w0 = __expf(t2) * aw;
    float w1 = __expf(t3) * ah;
    float ws = 2.0f - w0 * w1;
    float xd = obj * (t0 - pbx) * ws;
    float yd = obj * (t1 - pby) * ws;
    float wd = obj * (t2 - p2) * ws;
    float hd = obj * (t3 - p3) * ws;
    float pc = 1.0f / (1.0f + __expf(-p4));
    float cd = (-pc) + obj * (1.0f - pc) * 5.0f;  // NOOBJ=1, OBJ=5
    cd *= 1.0f - (1.0f - obj) * ign;
    float cld = obj * ce;

    if (lane == 0)
      acc += xd * xd + yd * yd + wd * wd + hd * hd + cd * cd + cld * cld;
  }

  if (lane == 0) atomicAdd(&out[b], acc);
}

extern "C" void kernel_launch(void* const* d_in, const int* in_sizes, int n_in,
                              void* d_out, int out_size, void* d_ws, size_t ws_size,
                              hipStream_t stream) {
  // d_in order: input_image (shape-only, unused), y_pred, y_true, true_boxes
  const float* y_pred = (const float*)d_in[1];
  const float* y_true = (const float*)d_in[2];
  const float* tboxes = (const float*)d_in[3];
  float* out = (float*)d_out;

  int B = in_sizes[2] / (CELLS_PER_BATCH * 85);
  if (B <= 0) B = 16;

  yolo_zero_kernel<<<1, 256, 0, stream>>>(out, out_size);
  dim3 grid(BLOCKS_PER_BATCH, B);
  yolo_loss_kernel<<<grid, 256, 0, stream>>>(y_pred, y_true, tboxes, out);
}